// RankingLoss_979252544202
// MI455X (gfx1250) — compile-verified
//
#include <hip/hip_runtime.h>

typedef __attribute__((ext_vector_type(2))) float v2f;
typedef __attribute__((ext_vector_type(8))) float v8f;

#define B_ROWS 16384
#define D_DIM  1024
#define C_CAT  14
#define WAVES  8
#define BLOCK  (WAVES * 32)

__global__ void rank_zero_out(float* out) { out[0] = 0.0f; }

__global__ __launch_bounds__(BLOCK)
void RankingLoss_979252544202_kernel(const float* __restrict__ zi,
                                     const float* __restrict__ zt,
                                     const int*   __restrict__ labels,
                                     float*       __restrict__ out) {
    __shared__ float Dtile[256];   // full 16x16 gram tile for this row group
    __shared__ float bnd[2];       // boundary dots: [0]=imp_img[15], [1]=imp_txt[15]
    __shared__ float blockSum;

    const int tid  = threadIdx.x;
    const int wave = tid >> 5;
    const int lane = tid & 31;
    const int g    = blockIdx.x;       // 16-row group index
    const int row0 = g * 16;

    if (tid < 256) Dtile[tid] = 0.0f;
    if (tid == 0) { bnd[0] = 0.0f; bnd[1] = 0.0f; blockSum = 0.0f; }
    __syncthreads();

    // ---- WMMA accumulation: D = A(img rows) x B(txt cols) over K chunk ----
    // f32 16x16x4 A layout: lanes 0-15 hold M=lane, VGPR0/1 = K0/K1;
    //                       lanes 16-31 hold M=lane-16, VGPR0/1 = K2/K3.
    const int m     = lane & 15;
    const int khalf = (lane >> 4) * 2;             // 0 or 2
    const float* arow = zi + (size_t)(row0 + m) * D_DIM;
    const float* brow = zt + (size_t)(row0 + m) * D_DIM;

    v8f acc = {};
    const int kBeg = wave * (D_DIM / WAVES);
    const int kEnd = kBeg + (D_DIM / WAVES);
#pragma unroll 8
    for (int k = kBeg; k < kEnd; k += 4) {
        v2f a = *(const v2f*)(arow + k + khalf);   // A[M][k+khalf .. +1]
        v2f b = *(const v2f*)(brow + k + khalf);   // B[k+khalf .. +1][N]
        // 8 args: (neg_a, A, neg_b, B, c_mod, C, reuse_a, reuse_b)
        acc = __builtin_amdgcn_wmma_f32_16x16x4_f32(
            false, a, false, b, (short)0, acc, false, false);
    }

    // Scatter partial C into LDS per documented C/D layout:
    //   VGPR r, lanes 0-15 -> (M=r,   N=lane); lanes 16-31 -> (M=r+8, N=lane-16)
    const int nIdx  = lane & 15;
    const int mBase = (lane >> 4) * 8;
#pragma unroll
    for (int r = 0; r < 8; ++r)
        atomicAdd(&Dtile[(mBase + r) * 16 + nIdx], acc[r]);

    // ---- boundary dot products for row row0+15 (needs row row0+16, wraps) ----
    {
        const size_t rcur  = (size_t)(row0 + 15) * D_DIM;
        const size_t rnext = (size_t)((row0 + 16) % B_ROWS) * D_DIM;
        float p0 = 0.0f, p1 = 0.0f;
        for (int k = tid; k < D_DIM; k += BLOCK) {
            p0 += zi[rnext + k] * zt[rcur + k];    // dot(img_{i+1}, txt_i) = imp_img[15]
            p1 += zt[rnext + k] * zi[rcur + k];    // dot(txt_{i+1}, img_i) = imp_txt[15]
        }
#pragma unroll
        for (int off = 16; off > 0; off >>= 1) {
            p0 += __shfl_down(p0, off, 32);
            p1 += __shfl_down(p1, off, 32);
        }
        if (lane == 0) { atomicAdd(&bnd[0], p0); atomicAdd(&bnd[1], p1); }
    }

    __syncthreads();

    // ---- per-row margin + hinge loss (threads 0..15, one row each) ----
    if (tid < 16) {
        const int i = row0 + tid;
        const int j = (i + 1) % B_ROWS;
        int eq = 1, nU = 0, dH = 0;
#pragma unroll
        for (int c = 0; c < C_CAT; ++c) {
            const int a = labels[(size_t)i * C_CAT + c];
            const int b = labels[(size_t)j * C_CAT + c];
            eq &= (a == b);
            nU += (a | b);
            dH += (a ^ b);
        }
        float margin;
        if (eq) {
            margin = 0.0f;
        } else {
            const float nf = (float)nU;
            const float df = (float)dH;
            margin = (nf > 0.0f) ? fmaxf(0.5f, df / fmaxf(nf, 1.0f)) : 0.5f;
        }
        const float paired = Dtile[tid * 16 + tid];
        const float impTxt = (tid < 15) ? Dtile[tid * 16 + tid + 1] : bnd[1];
        const float impImg = (tid < 15) ? Dtile[(tid + 1) * 16 + tid] : bnd[0];
        const float v = fmaxf(impImg - paired + margin, 0.0f)
                      + fmaxf(impTxt - paired + margin, 0.0f);
        atomicAdd(&blockSum, v);
    }
    __syncthreads();

    if (tid == 0)
        atomicAdd(out, blockSum * (1.0f / (float)B_ROWS));
}

extern "C" void kernel_launch(void* const* d_in, const int* in_sizes, int n_in,
                              void* d_out, int out_size, void* d_ws, size_t ws_size,
                              hipStream_t stream) {
    const float* zi     = (const float*)d_in[0];
    const float* zt     = (const float*)d_in[1];
    const int*   labels = (const int*)d_in[2];
    float* out = (float*)d_out;

    rank_zero_out<<<1, 1, 0, stream>>>(out);
    RankingLoss_979252544202_kernel<<<B_ROWS / 16, BLOCK, 0, stream>>>(zi, zt, labels, out);
}